// GMNet_16346645529218
// MI455X (gfx1250) — compile-verified
//
#include <hip/hip_runtime.h>
#include <hip/hip_bf16.h>
#include <math.h>

typedef __attribute__((ext_vector_type(2))) float v2f;
typedef __attribute__((ext_vector_type(8))) float v8f;

// ---------------------------------------------------------------------------
// C[m][n] = sum_k A[m][k]*B[n][k]  (+bias[n], relu)   -- fp32 WMMA 16x16x4
// grid: (M/16, N/16), block: 32 (one wave per 16x16 tile)
// ---------------------------------------------------------------------------
__global__ __launch_bounds__(32) void gemm_nt_wmma(
    const float* __restrict__ A, int lda,
    const float* __restrict__ B, int ldb,
    float* __restrict__ C, int ldc, int K,
    const float* __restrict__ bias, int relu)
{
    const int lane = threadIdx.x;
    const int half = lane >> 4;   // 0: K pair {0,1}, 1: K pair {2,3}
    const int l16  = lane & 15;
    const int m0 = blockIdx.x * 16;
    const int n0 = blockIdx.y * 16;
    const float* arow = A + (size_t)(m0 + l16) * lda;  // A lane holds row m = l16
    const float* brow = B + (size_t)(n0 + l16) * ldb;  // B^T lane holds col n = l16
    __builtin_prefetch(arow, 0, 1);
    __builtin_prefetch(brow, 0, 1);
    v8f acc = {};
    for (int k = 0; k < K; k += 4) {
        const int kk = k + 2 * half;
        v2f av = { arow[kk], arow[kk + 1] };
        v2f bv = { brow[kk], brow[kk + 1] };
        acc = __builtin_amdgcn_wmma_f32_16x16x4_f32(
            false, av, false, bv, (short)0, acc, false, false);
    }
    const float bb = bias ? bias[n0 + l16] : 0.0f;
#pragma unroll
    for (int r = 0; r < 8; ++r) {
        const int m = m0 + r + 8 * half;      // C: vgpr r -> M=r (lanes 0-15), M=r+8 (16-31)
        float v = acc[r] + bb;
        if (relu) v = fmaxf(v, 0.0f);
        C[(size_t)m * ldc + n0 + l16] = v;
    }
}

// ---------------------------------------------------------------------------
// C[m][n] = sum_k A[m][k]*B[k][n]   -- fp32 WMMA 16x16x4
// ---------------------------------------------------------------------------
__global__ __launch_bounds__(32) void gemm_nn_wmma(
    const float* __restrict__ A, int lda,
    const float* __restrict__ B, int ldb,
    float* __restrict__ C, int ldc, int K)
{
    const int lane = threadIdx.x;
    const int half = lane >> 4;
    const int l16  = lane & 15;
    const int m0 = blockIdx.x * 16;
    const int n0 = blockIdx.y * 16;
    const float* arow = A + (size_t)(m0 + l16) * lda;
    __builtin_prefetch(arow, 0, 1);
    v8f acc = {};
    for (int k = 0; k < K; k += 4) {
        const int kk = k + 2 * half;
        v2f av = { arow[kk], arow[kk + 1] };
        v2f bv = { B[(size_t)kk * ldb + n0 + l16],
                   B[(size_t)(kk + 1) * ldb + n0 + l16] };
        acc = __builtin_amdgcn_wmma_f32_16x16x4_f32(
            false, av, false, bv, (short)0, acc, false, false);
    }
#pragma unroll
    for (int r = 0; r < 8; ++r) {
        const int m = m0 + r + 8 * half;
        C[(size_t)m * ldc + n0 + l16] = acc[r];
    }
}

__global__ void transpose80(const float* __restrict__ in, float* __restrict__ out)
{
    int i = blockIdx.x * blockDim.x + threadIdx.x;
    if (i < 6400) { int r = i / 80, c = i - r * 80; out[c * 80 + r] = in[i]; }
}

// x[row] = data[row] + (||data[row]|| / ||m[row]||) * m[row]     (d = 512)
__global__ __launch_bounds__(256) void lam_combine(
    const float* __restrict__ d, const float* __restrict__ m, float* __restrict__ x)
{
    __shared__ float s1[256], s2[256];
    const int row = blockIdx.x, t = threadIdx.x;
    const float* dr = d + (size_t)row * 512;
    const float* mr = m + (size_t)row * 512;
    float a = 0.f, b = 0.f;
    for (int k = t; k < 512; k += 256) { float u = dr[k], v = mr[k]; a += u * u; b += v * v; }
    s1[t] = a; s2[t] = b; __syncthreads();
    for (int s = 128; s > 0; s >>= 1) {
        if (t < s) { s1[t] += s1[t + s]; s2[t] += s2[t + s]; }
        __syncthreads();
    }
    const float lam = sqrtf(s1[0]) / sqrtf(s2[0]);
    float* xr = x + (size_t)row * 512;
    for (int k = t; k < 512; k += 256) xr[k] = dr[k] + lam * mr[k];
}

// e[row] /= max(||e[row]||, 1e-12)    (d = 512)
__global__ __launch_bounds__(256) void l2norm_rows(float* __restrict__ e)
{
    __shared__ float s1[256];
    const int row = blockIdx.x, t = threadIdx.x;
    float* er = e + (size_t)row * 512;
    float a = 0.f;
    for (int k = t; k < 512; k += 256) { float u = er[k]; a += u * u; }
    s1[t] = a; __syncthreads();
    for (int s = 128; s > 0; s >>= 1) { if (t < s) s1[t] += s1[t + s]; __syncthreads(); }
    const float inv = 1.0f / fmaxf(sqrtf(s1[0]), 1e-12f);
    for (int k = t; k < 512; k += 256) er[k] *= inv;
}

// ---------------------------------------------------------------------------
// Persistent single-workgroup structured KKT solver.
//   Q = kappa*I - Mmat, Mmat(v) computed in O(n^2) from P (inclusion-exclusion)
//   Q^{-1} via Neumann (16 sweeps), constraints via Schur PCG with analytic
//   (A A^T)^{-1} preconditioner, then clip + softmax(1000*) epilogue.
// ---------------------------------------------------------------------------
#define NT 1024
__global__ __launch_bounds__(1024) void solver_kernel(
    const float* __restrict__ Pg,
    float* __restrict__ bvec, float* __restrict__ xv, float* __restrict__ tv,
    float* __restrict__ out)
{
    const int tid = threadIdx.x;
    __shared__ float Pl[6400];                        // P[i1*80+i2]
    __shared__ float Rr[80], Cc[80], D1[80], D2[80];  // structured matvec partials
    __shared__ float ST[2];                           // S = sum v, T = sum P.v
    __shared__ float lamv[159], rv[159], hv[159], prv[159], qv[159];
    __shared__ float scal[6];                         // 0:rho 1:dot 2:alpha 3:beta 4:sx 5:sy

    for (int i = tid; i < 6400; i += NT) Pl[i] = Pg[i];
    if (tid < 159) lamv[tid] = 0.0f;
    __syncthreads();

    const float kap  = 6241.0f + 1e-6f;   // (n1-1)(n2-1) + ridge
    const float invk = 1.0f / kap;

    // reductions for Mmat*src
    auto reduceMV = [&](const float* src) {
        if (tid < 80) { Rr[tid] = 0.f; Cc[tid] = 0.f; D1[tid] = 0.f; D2[tid] = 0.f; }
        if (tid == 0) { ST[0] = 0.f; ST[1] = 0.f; }
        __syncthreads();
        float sS = 0.f, sT = 0.f;
        for (int e = tid; e < 6400; e += NT) {
            const int i2 = e / 80, i1 = e - i2 * 80;
            const float v  = src[e];
            const float Pi = Pl[i1 * 80 + i2];
            atomicAdd(&Rr[i1], v);      atomicAdd(&Cc[i2], v);
            atomicAdd(&D1[i1], Pi * v); atomicAdd(&D2[i2], Pi * v);
            sS += v; sT += Pi * v;
        }
        for (int o = 16; o > 0; o >>= 1) {
            sS += __shfl_xor(sS, o, 32);
            sT += __shfl_xor(sT, o, 32);
        }
        if ((tid & 31) == 0) { atomicAdd(&ST[0], sS); atomicAdd(&ST[1], sT); }
        __syncthreads();
    };

    // dst = (bsrc + Mmat*src) / kappa     (one Neumann sweep)
    auto sweep = [&](const float* bsrc, const float* src, float* dst) {
        reduceMV(src);
        const float S = ST[0], T = ST[1];
        for (int e = tid; e < 6400; e += NT) {
            const int i2 = e / 80, i1 = e - i2 * 80;
            const float v  = src[e];
            const float Pi = Pl[i1 * 80 + i2];
            const float mx = 0.5f * Pi * (S - Rr[i1] - Cc[i2] + v)
                           + 0.5f * (T - D1[i1] - D2[i2] + Pi * v);
            dst[e] = (bsrc[e] + mx) * invk;
        }
        __syncthreads();
    };

    // xv = Q^{-1} bvec  (Neumann, 16 sweeps: rho(M/kappa) ~ 0.1 -> fp32-converged)
    auto solveQ = [&]() {
        for (int e = tid; e < 6400; e += NT) xv[e] = bvec[e] * invk;
        __syncthreads();
        float* cur = xv; float* nxt = tv;
        for (int j = 0; j < 16; ++j) { sweep(bvec, cur, nxt); float* t2 = cur; cur = nxt; nxt = t2; }
        // 16 (even) iterations -> result resides in xv
    };

    // dstC[0..158] = A*src  (row sums over i2, then over i1; last row dropped)
    auto applyA = [&](const float* src, float* dstC) {
        if (tid < 80) { Rr[tid] = 0.f; Cc[tid] = 0.f; }
        __syncthreads();
        for (int e = tid; e < 6400; e += NT) {
            const int i2 = e / 80, i1 = e - i2 * 80;
            const float v = src[e];
            atomicAdd(&Rr[i1], v); atomicAdd(&Cc[i2], v);
        }
        __syncthreads();
        if (tid < 159) dstC[tid] = (tid < 80) ? Rr[tid] : Cc[tid - 80];
        __syncthreads();
    };

    // bvec = coefP * p + sign * (A^T lam),   p[(i2,i1)] = P[i1,i2]
    auto setB = [&](const float* lam, float coefP, float sign) {
        for (int e = tid; e < 6400; e += NT) {
            const int i2 = e / 80, i1 = e - i2 * 80;
            const float at = lam[i1] + ((i2 < 79) ? lam[80 + i2] : 0.0f);
            bvec[e] = coefP * Pl[i1 * 80 + i2] + sign * at;
        }
        __syncthreads();
    };

    // hv = kappa * (A A^T)^{-1} rv  — closed form: AA^T=[[80I,J],[J^T,80I_79]]
    auto prec = [&]() {
        if (tid == 0) {
            float sa = 0.f, sb = 0.f;
            for (int k = 0; k < 80; ++k)  sa += rv[k];
            for (int k = 80; k < 159; ++k) sb += rv[k];
            const float sx = sa - sb;
            const float sy = sa / 80.0f - sx;
            scal[4] = sx; scal[5] = sy;
        }
        __syncthreads();
        if (tid < 159) {
            const float s = (tid < 80) ? scal[5] : scal[4];
            hv[tid] = kap * (rv[tid] - s) * (1.0f / 80.0f);
        }
        __syncthreads();
    };

    auto dot159 = [&](const float* a, const float* b, int slot) {
        if (tid == 0) { float s = 0.f; for (int k = 0; k < 159; ++k) s += a[k] * b[k]; scal[slot] = s; }
        __syncthreads();
    };

    // ---- Schur rhs: g = A Q^{-1} p - 1
    setB(lamv, 1.0f, 0.0f);
    solveQ();
    applyA(xv, qv);
    if (tid < 159) rv[tid] = qv[tid] - 1.0f;
    __syncthreads();
    prec();
    if (tid < 159) prv[tid] = hv[tid];
    __syncthreads();
    dot159(rv, hv, 0);                       // rho

    // ---- PCG on  (A Q^{-1} A^T) lam = g
    for (int it = 0; it < 12; ++it) {
        setB(prv, 0.0f, 1.0f);               // bvec = A^T pr
        solveQ();
        applyA(xv, qv);                      // q = S * pr
        dot159(prv, qv, 1);
        if (tid == 0) scal[2] = scal[0] / (scal[1] + 1e-30f);   // alpha
        __syncthreads();
        if (tid < 159) { lamv[tid] += scal[2] * prv[tid]; rv[tid] -= scal[2] * qv[tid]; }
        __syncthreads();
        prec();
        dot159(rv, hv, 1);                   // rho_new
        if (tid == 0) { scal[3] = scal[1] / (scal[0] + 1e-30f); scal[0] = scal[1]; }
        __syncthreads();
        if (tid < 159) prv[tid] = hv[tid] + scal[3] * prv[tid];
        __syncthreads();
    }

    // ---- z = Q^{-1}(p - A^T lam)   -> xv
    setB(lamv, 1.0f, -1.0f);
    solveQ();

    // ---- epilogue: s[i1][i2] = clip(z[i2*80+i1]); out = softmax(1000*s, axis i2)
    const int w = tid >> 5, l = tid & 31;
    for (int i1 = w; i1 < 80; i1 += 32) {
        float vloc[3], eloc[3], mx = -1e30f;
#pragma unroll
        for (int jj = 0; jj < 3; ++jj) {
            const int i2 = l + jj * 32;
            float v = -1e30f;
            if (i2 < 80) v = fminf(fmaxf(xv[i2 * 80 + i1], 0.0f), 1.0f);
            vloc[jj] = v; mx = fmaxf(mx, v);
        }
        for (int o = 16; o > 0; o >>= 1) mx = fmaxf(mx, __shfl_xor(mx, o, 32));
        float se = 0.f;
#pragma unroll
        for (int jj = 0; jj < 3; ++jj) {
            const int i2 = l + jj * 32;
            float ev = 0.0f;
            if (i2 < 80) ev = __expf(1000.0f * (vloc[jj] - mx));
            eloc[jj] = ev; se += ev;
        }
        for (int o = 16; o > 0; o >>= 1) se += __shfl_xor(se, o, 32);
#pragma unroll
        for (int jj = 0; jj < 3; ++jj) {
            const int i2 = l + jj * 32;
            if (i2 < 80) out[i1 * 80 + i2] = eloc[jj] / se;
        }
    }
}

// ---------------------------------------------------------------------------
extern "C" void kernel_launch(void* const* d_in, const int* in_sizes, int n_in,
                              void* d_out, int out_size, void* d_ws, size_t ws_size,
                              hipStream_t stream)
{
    (void)in_sizes; (void)n_in; (void)out_size; (void)ws_size;
    const float* data1 = (const float*)d_in[0];   // 80x512
    const float* data2 = (const float*)d_in[1];   // 80x512
    const float* W     = (const float*)d_in[2];   // 512x512
    const float* bias  = (const float*)d_in[3];   // 512
    float* out = (float*)d_out;                   // 1x80x80

    float* ws  = (float*)d_ws;
    float* P0  = ws;              // 6400
    float* P0T = P0  + 6400;      // 6400
    float* m1  = P0T + 6400;      // 40960
    float* m2  = m1  + 40960;     // 40960
    float* x1  = m2  + 40960;     // 40960
    float* x2  = x1  + 40960;     // 40960
    float* e1  = x2  + 40960;     // 40960
    float* e2  = e1  + 40960;     // 40960
    float* P   = e2  + 40960;     // 6400
    float* bv  = P   + 6400;      // 6400
    float* xvv = bv  + 6400;      // 6400
    float* tvv = xvv + 6400;      // 6400   (total ~1.08 MiB)

    const dim3 b32(32);
    // P0 = data1 @ data2^T
    gemm_nt_wmma<<<dim3(5, 5),  b32, 0, stream>>>(data1, 512, data2, 512, P0, 80, 512, nullptr, 0);
    transpose80<<<25, 256, 0, stream>>>(P0, P0T);
    // m1 = P0 @ data2 ; m2 = P0^T @ data1
    gemm_nn_wmma<<<dim3(5, 32), b32, 0, stream>>>(P0,  80, data2, 512, m1, 512, 80);
    gemm_nn_wmma<<<dim3(5, 32), b32, 0, stream>>>(P0T, 80, data1, 512, m2, 512, 80);
    // x = data + lam * m
    lam_combine<<<80, 256, 0, stream>>>(data1, m1, x1);
    lam_combine<<<80, 256, 0, stream>>>(data2, m2, x2);
    // e = relu(x @ W^T + b), row-normalized
    gemm_nt_wmma<<<dim3(5, 32), b32, 0, stream>>>(x1, 512, W, 512, e1, 512, 512, bias, 1);
    gemm_nt_wmma<<<dim3(5, 32), b32, 0, stream>>>(x2, 512, W, 512, e2, 512, 512, bias, 1);
    l2norm_rows<<<80, 256, 0, stream>>>(e1);
    l2norm_rows<<<80, 256, 0, stream>>>(e2);
    // P = e1 @ e2^T  (drives both Mmat and the QP linear term)
    gemm_nt_wmma<<<dim3(5, 5),  b32, 0, stream>>>(e1, 512, e2, 512, P, 80, 512, nullptr, 0);
    // structured KKT solve + clip + softmax epilogue (persistent, LDS-resident)
    solver_kernel<<<1, 1024, 0, stream>>>(P, bv, xvv, tvv, out);
}